// CoreMLAlignmentStandalone_83202106458515
// MI455X (gfx1250) — compile-verified
//
#include <hip/hip_runtime.h>
#include <hip/hip_bf16.h>
#include <stdint.h>

// Problem constants (fixed shapes from the reference).
#define NTOK 2048            // L: tokens
#define NFRM 8192            // F: MAX_FRAMES
#define CD   640             // channels of d
#define CT   512             // channels of t_en
#define EN_ELEMS  (CD * NFRM)   // 5,242,880
#define ASR_ELEMS (CT * NFRM)   // 4,194,304

// ---------------------------------------------------------------------------
// Kernel 1: build frame->token map from durations.
// Single 256-thread block. Stages pred_dur (8 KB) into LDS via the gfx1250
// async global->LDS DMA path (ASYNCcnt-tracked), then does a block scan.
// map[f] = token index owning frame f, or -1 for f >= total duration.
// ---------------------------------------------------------------------------
__global__ __launch_bounds__(256) void build_map_kernel(const int* __restrict__ pred_dur,
                                                        int* __restrict__ map) {
    __shared__ __align__(16) int sdur[NTOK];
    __shared__ int spart[256];
    const int t = threadIdx.x;

    // --- async global -> LDS copy of 2048 ints (2 passes of 256 lanes x 16B) ---
    {
        // Low 32 bits of the generic pointer to LDS == wave-relative LDS byte
        // address (LDS aperture uses addr[31:0]).
        uint32_t lds_base = (uint32_t)(uintptr_t)(&sdur[0]);
        uint64_t gbase    = (uint64_t)(uintptr_t)pred_dur;
        #pragma unroll
        for (int pass = 0; pass < 2; ++pass) {
            uint32_t byteoff = (uint32_t)(pass * 256 + t) * 16u;
            uint32_t laddr   = lds_base + byteoff;
            uint64_t gaddr   = gbase + byteoff;
            asm volatile("global_load_async_to_lds_b128 %0, %1, off"
                         :: "v"(laddr), "v"(gaddr)
                         : "memory");
        }
        // Wait for this wave's async copies; barrier below syncs the 8 waves.
        asm volatile("s_wait_asynccnt 0" ::: "memory");
    }
    __syncthreads();

    // --- per-thread partial sums over 8 tokens each ---
    int mydur[8];
    int sum = 0;
    #pragma unroll
    for (int j = 0; j < 8; ++j) { mydur[j] = sdur[t * 8 + j]; sum += mydur[j]; }
    spart[t] = sum;
    __syncthreads();

    // --- Hillis-Steele inclusive scan over 256 partials ---
    for (int off = 1; off < 256; off <<= 1) {
        int v   = spart[t];
        int add = (t >= off) ? spart[t - off] : 0;
        __syncthreads();
        spart[t] = v + add;
        __syncthreads();
    }
    const int excl  = (t == 0) ? 0 : spart[t - 1];
    const int total = spart[255];        // T_a <= 2048*4 = 8192 = NFRM

    // Frames past T_a map to "no token". Disjoint from the fill below.
    for (int f = total + t; f < NFRM; f += 256) map[f] = -1;

    // Fill each token's contiguous frame range (dur in [0,4]).
    int run = excl;
    #pragma unroll
    for (int j = 0; j < 8; ++j) {
        const int l = t * 8 + j;
        const int e = run + mydur[j];
        for (int f = run; f < e; ++f) map[f] = l;
        run = e;
    }
}

// ---------------------------------------------------------------------------
// Kernel 2: coalesced scatter/gather producing both outputs.
//   en [640 x 8192] then asr [512 x 8192], concatenated flat in d_out.
// One float4 of output per thread; writes perfectly coalesced along f.
// ---------------------------------------------------------------------------
__global__ __launch_bounds__(256) void scatter_kernel(const float* __restrict__ d,
                                                      const float* __restrict__ t_en,
                                                      const int* __restrict__ map,
                                                      float* __restrict__ out) {
    const long long idx = (long long)blockIdx.x * blockDim.x + threadIdx.x;
    const long long e4  = idx * 4;

    if (e4 < (long long)EN_ELEMS) {
        // en[c, f] = d[tok(f), c]
        const int c = (int)(e4 >> 13);       // / 8192
        const int f = (int)(e4 & 8191);      // % 8192, multiple of 4
        const int4 tok = *(const int4*)(map + f);
        const float* __restrict__ dc = d + c;
        float4 v;
        v.x = (tok.x >= 0) ? dc[(size_t)tok.x * CD] : 0.0f;
        v.y = (tok.y >= 0) ? dc[(size_t)tok.y * CD] : 0.0f;
        v.z = (tok.z >= 0) ? dc[(size_t)tok.z * CD] : 0.0f;
        v.w = (tok.w >= 0) ? dc[(size_t)tok.w * CD] : 0.0f;
        *(float4*)(out + e4) = v;
    } else {
        // asr[c, f] = t_en[c, tok(f)]
        const long long e = e4 - EN_ELEMS;
        const int c = (int)(e >> 13);
        const int f = (int)(e & 8191);
        const int4 tok = *(const int4*)(map + f);
        const float* __restrict__ tc = t_en + (size_t)c * NTOK;
        float4 v;
        v.x = (tok.x >= 0) ? tc[tok.x] : 0.0f;
        v.y = (tok.y >= 0) ? tc[tok.y] : 0.0f;
        v.z = (tok.z >= 0) ? tc[tok.z] : 0.0f;
        v.w = (tok.w >= 0) ? tc[tok.w] : 0.0f;
        *(float4*)(out + (long long)EN_ELEMS + e) = v;
    }
}

extern "C" void kernel_launch(void* const* d_in, const int* in_sizes, int n_in,
                              void* d_out, int out_size, void* d_ws, size_t ws_size,
                              hipStream_t stream) {
    const int*   pred_dur = (const int*)d_in[0];    // [1, 2048] int32
    const float* d        = (const float*)d_in[1];  // [1, 2048, 640] f32
    const float* t_en     = (const float*)d_in[2];  // [1, 512, 2048] f32
    float*       out      = (float*)d_out;          // en (640*8192) ++ asr (512*8192)
    int*         map      = (int*)d_ws;             // 8192 ints = 32 KB scratch

    build_map_kernel<<<1, 256, 0, stream>>>(pred_dur, map);

    const long long total4 = ((long long)EN_ELEMS + ASR_ELEMS) / 4;  // 2,359,296
    const int blocks = (int)(total4 / 256);                          // 9,216 exact
    scatter_kernel<<<blocks, 256, 0, stream>>>(d, t_en, map, out);
}